// AllZeroDigitalFilter_69234872811592
// MI455X (gfx1250) — compile-verified
//
#include <hip/hip_runtime.h>

// ---------------------------------------------------------------------------
// Time-varying FIR via fp32 WMMA (V_WMMA_F32_16X16X4_F32) on gfx1250.
//
// y[t] = (1-f)*U[t,k] + f*U[t,k+1],  k = t/P, f = (t%P)/P,
// U[t,m] = sum_j h[m,j] * x[t-j]  (fixed-coefficient convolution -> GEMM).
//
// Per 16-sample tile:  D(16x16) = A(16xK) * B(Kx16),  K = 52 (50 taps + 2 zero pad)
//   A[m,j] = h[k0+m, j]        (frame-major coefficient block, wave-invariant)
//   B[j,n] = x[t0 + n - j]     (Hankel slice of the staged x window)
//   D[m,n] = U[t0+n, k0+m]
//
// x window is staged REVERSED in LDS so each B fragment is an ascending
// adjacent pair (ds_load_2addr_b32 in-order -> no register-swap movs).
// ---------------------------------------------------------------------------

#define T_LEN   80000
#define BATCH   16
#define P_FR    80
#define NFR     1000          // frames = T/P
#define TAPS    50            // M+1
#define KPAD    52            // taps padded to multiple of 4 (13 chunks)
#define KCHUNKS 13
#define FR_BLK  8             // frames per workgroup
#define SPAN    (FR_BLK * P_FR)          // 640 samples per workgroup
#define BLKS_PB (NFR / FR_BLK)           // 125 blocks per batch
#define XWIN    (2 + 49 + SPAN)          // 691 = guard(2) + history(49) + span
#define HROWS   16
#define NWAVES  8
#define TILES_PER_WAVE (SPAN / (NWAVES * 16))  // 5

typedef float v2f __attribute__((ext_vector_type(2)));
typedef float v8f __attribute__((ext_vector_type(8)));

__global__ __launch_bounds__(256)
void tvfir_wmma_f32_kernel(const float* __restrict__ x,
                           const float* __restrict__ h,
                           float* __restrict__ y) {
    __shared__ float Xr[704];                 // REVERSED x window (691 used)
    __shared__ float Hl[HROWS * KPAD];        // 16 frames x 52 taps (zero padded)
    __shared__ float Dl[NWAVES][16 * 16];     // per-wave D-tile staging

    const int blk    = blockIdx.x;
    const int b      = blk / BLKS_PB;
    const int fb     = blk % BLKS_PB;
    const int k0     = fb * FR_BLK;           // first frame of this block
    const int t_base = k0 * P_FR;             // first output sample of this block
    const int tid    = threadIdx.x;

    // ---- Stage reversed x window: Xr[q] = x[t_base + 639 - q] ----
    // q in [0,689): payload (zero-filled where t<0); q=689,690: guard zeros
    // backing the two zero-padded taps.
    for (int q = tid; q < XWIN; q += 256) {
        const int t = t_base + (SPAN - 1) - q;     // SPAN-1 = 639
        float v = 0.0f;
        if (q <= 688 && t >= 0) v = x[(size_t)b * T_LEN + t];
        Xr[q] = v;
    }
    // ---- Stage coefficient block: frames k0..k0+15 (clamped = h_pad replicate) ----
    for (int i = tid; i < HROWS * KPAD; i += 256) {
        const int row = i / KPAD;
        const int tap = i % KPAD;
        int fr = k0 + row;
        if (fr > NFR - 1) fr = NFR - 1;            // replicate last frame (h_pad)
        float v = 0.0f;
        if (tap < TAPS) v = h[((size_t)b * NFR + fr) * TAPS + tap];
        Hl[i] = v;
    }
    __syncthreads();

    const int wv   = tid >> 5;
    const int lane = tid & 31;
    const int nrow = lane & 15;               // A row (frame) / B-D column (n)
    const int hi   = lane >> 4;               // lane half select

    // ---- Preload A fragments once: invariant over all tiles of this block ----
    // ISA 16x4 fp32 A layout: lanes 0-15 hold (K=4c, 4c+1), lanes 16-31 (4c+2, 4c+3).
    // tap is always even -> these lower to 8B-aligned ds_load_b64.
    v2f afrag[KCHUNKS];
#pragma unroll
    for (int c = 0; c < KCHUNKS; ++c) {
        const int tap = 4 * c + 2 * hi;
        afrag[c][0] = Hl[nrow * KPAD + tap];
        afrag[c][1] = Hl[nrow * KPAD + tap + 1];
    }

    float* Dw = &Dl[wv][0];

    for (int tile = 0; tile < TILES_PER_WAVE; ++tile) {
        const int t0   = t_base + (wv * TILES_PER_WAVE + tile) * 16;
        const int toff = t0 - t_base;
        // Reversed index of tap (4c + 2*hi) for column nrow:
        const int qb = (SPAN - 1) - toff - nrow + 2 * hi;   // [0, 641]

        v8f acc = {};
#pragma unroll
        for (int c = 0; c < KCHUNKS; ++c) {
            v2f bfrag;
            bfrag[0] = Xr[qb + 4 * c];          // tap 4c + 2*hi
            bfrag[1] = Xr[qb + 4 * c + 1];      // tap 4c + 2*hi + 1  (ascending pair)
            acc = __builtin_amdgcn_wmma_f32_16x16x4_f32(
                false, afrag[c], false, bfrag, (short)0, acc, false, false);
        }

        // ---- Spill D tile to LDS: VGPR r, lane l -> row r + 8*hi, col l%16 ----
#pragma unroll
        for (int r = 0; r < 8; ++r) {
            Dw[(r + 8 * hi) * 16 + nrow] = acc[r];
        }

        // ---- Blend U[t,k] and U[t,k+1]; same-wave DS ordering, no barrier ----
        if (lane < 16) {
            const int   t = t0 + lane;
            const int   k = t / P_FR;
            const float f = (float)(t % P_FR) * (1.0f / P_FR);
            const int   m = k - k0;               // 0..7 ; m+1 <= 8 < 16 rows
            const float u = Dw[m * 16 + lane];
            const float v = Dw[(m + 1) * 16 + lane];
            y[(size_t)b * T_LEN + t] = (1.0f - f) * u + f * v;
        }
    }
}

extern "C" void kernel_launch(void* const* d_in, const int* in_sizes, int n_in,
                              void* d_out, int out_size, void* d_ws, size_t ws_size,
                              hipStream_t stream) {
    (void)in_sizes; (void)n_in; (void)d_ws; (void)ws_size; (void)out_size;
    const float* x = (const float*)d_in[0];   // (16, 80000) fp32
    const float* h = (const float*)d_in[1];   // (16, 1000, 50) fp32
    float*       y = (float*)d_out;           // (16, 80000) fp32

    dim3 grid(BATCH * BLKS_PB);               // 2000 workgroups
    dim3 block(256);                          // 8 wave32 per workgroup
    tvfir_wmma_f32_kernel<<<grid, block, 0, stream>>>(x, h, y);
}